// ConvKernel_24713241821292
// MI455X (gfx1250) — compile-verified
//
#include <hip/hip_runtime.h>

// Quantum CNOT-ring circuit, algebraically collapsed (see round-0 derivation):
//   out[b] = cpe*bqe + cpo*bqo
// where cpe/cpo are parity-masked sums of |C[p]|^2 = |r00*A[p] + r01*A[p^12]|^2
// over the high-nibble half-kron A, and bqe/bqo are parity-masked sums of
// |B[q]|^2 over the low-nibble half-kron B.
//
// The four 16-term masked reductions per element are computed on the matrix
// pipe: V_WMMA_F32_16X16X4_F32 with a constant 0/1 mask matrix as A-operand
// (rows 0..3 = the four parity classes) and per-lane magnitudes as B-operand.

typedef float v2f __attribute__((ext_vector_type(2)));
typedef float v8f __attribute__((ext_vector_type(8)));

struct C2 { float re, im; };

__device__ __forceinline__ C2 cmul(const C2 a, const C2 b) {
  return C2{ a.re * b.re - a.im * b.im, a.re * b.im + a.im * b.re };
}

// Row (parity class) that concatenated k-index 0..31 belongs to:
//   k<16 : |C[p]|^2 values, class = parity(p&7)        -> rows 0 (even) / 1 (odd)
//   k>=16: |B[q]|^2 values, class = parity(q), q=k-16  -> rows 2 (even) / 3 (odd)
__host__ __device__ constexpr int mrow(int k) {
  return (k < 16) ? ((k ^ (k >> 1) ^ (k >> 2)) & 1)
                  : (2 + ((k ^ (k >> 1) ^ (k >> 2) ^ (k >> 3)) & 1));
}

// Symmetric half-wave rotation: lane l receives lane (l+16)&31's value.
__device__ __forceinline__ float rot16f(int lane, float x) {
  const int idx = ((lane + 16) & 31) << 2;
  return __int_as_float(__builtin_amdgcn_ds_bpermute(idx, __float_as_int(x)));
}

__global__ __launch_bounds__(256) void qring_kernel(
    const float* __restrict__ in,     // (batch, 8) f32
    const float* __restrict__ wre,    // (3,) f32
    const float* __restrict__ wim,    // (3,) f32
    float* __restrict__ out,          // (batch,) f32
    int batch)
{
  const int b    = blockIdx.x * 256 + threadIdx.x;
  const int lane = threadIdx.x & 31;          // wave32
  const bool lo16 = (lane < 16);
  const int mrid  = lane & 15;                // A-matrix row M carried by this lane
  if (b >= batch) return;                     // batch % 256 == 0: no divergence

  // ---- uniform rotation: rot = R(w2)R(w1)R(w0) = R(W), W = (S_re + i S_im)/2
  // r00 = cos(W), r01 = -sin(W);  cos(x+iy)=cx*chy - i sx*shy, sin(x+iy)=sx*chy + i cx*shy
  const float X = 0.5f * (wre[0] + wre[1] + wre[2]);
  const float Y = 0.5f * (wim[0] + wim[1] + wim[2]);
  float sX, cX;
  __sincosf(X, &sX, &cX);
  const float eY  = __expf(Y);
  const float rY  = 1.0f / eY;
  const float chY = 0.5f * (eY + rY);
  const float shY = 0.5f * (eY - rY);
  const C2 r00{  cX * chY, -sX * shY };
  const C2 r01{ -sX * chY, -cX * shY };

  // ---- coalesced vector load of this element's 8 inputs (2x b128)
  const float4* ip = reinterpret_cast<const float4*>(in) + (size_t)b * 2;
  const float4 xlo = ip[0];
  const float4 xhi = ip[1];
  const float x[8] = { xlo.x, xlo.y, xlo.z, xlo.w, xhi.x, xhi.y, xhi.z, xhi.w };

  // ---- per-qubit amplitudes: a0 = cos(ry) e^{-i rz}, a1 = sin(ry) e^{+i rz}
  // ry, rz in (-pi/4, pi/4): native v_sin/v_cos are safe, no range reduction.
  C2 amp[8][2];
#pragma unroll
  for (int q = 0; q < 8; ++q) {
    const float xv = x[q];
    const float ry = 0.5f * atanf(xv);
    const float rz = 0.5f * atanf(xv * xv);
    float sy, cy, sz, cz;
    __sincosf(ry, &sy, &cy);
    __sincosf(rz, &sz, &cz);
    amp[q][0] = C2{ cy * cz, -cy * sz };
    amp[q][1] = C2{ sy * cz,  sy * sz };
  }

  // ---- concatenated magnitude vector: val[0..15]=|C[p]|^2, val[16..31]=|B[q]|^2
  float val[32];
  {
    C2 t4[4], t8[8], A[16];
#pragma unroll
    for (int i = 0; i < 4; ++i)  t4[i] = cmul(amp[0][(i >> 1) & 1], amp[1][i & 1]);
#pragma unroll
    for (int i = 0; i < 8; ++i)  t8[i] = cmul(t4[i >> 1], amp[2][i & 1]);
#pragma unroll
    for (int i = 0; i < 16; ++i) A[i] = cmul(t8[i >> 1], amp[3][i & 1]);
#pragma unroll
    for (int p = 0; p < 16; ++p) {
      const C2 a0 = A[p];
      const C2 a1 = A[p ^ 12];
      const float cre = r00.re * a0.re - r00.im * a0.im + r01.re * a1.re - r01.im * a1.im;
      const float cim = r00.re * a0.im + r00.im * a0.re + r01.re * a1.im + r01.im * a1.re;
      val[p] = cre * cre + cim * cim;
    }
#pragma unroll
    for (int i = 0; i < 4; ++i)  t4[i] = cmul(amp[4][(i >> 1) & 1], amp[5][i & 1]);
#pragma unroll
    for (int i = 0; i < 8; ++i)  t8[i] = cmul(t4[i >> 1], amp[6][i & 1]);
#pragma unroll
    for (int i = 0; i < 16; ++i) {
      const C2 Bq = cmul(t8[i >> 1], amp[7][i & 1]);
      val[16 + i] = Bq.re * Bq.re + Bq.im * Bq.im;
    }
  }

  // ---- masked reductions on the matrix pipe: D = MASK(16x4) * VAL(4x16) + D
  // A-operand layout (f32 16x4): lanes 0-15 M=0-15 carry K=k0,k0+1 in the two
  // VGPRs; lanes 16-31 carry K=k0+2,k0+3.  B-operand mirrors it with N=lane%16.
  // Pass 1 reduces elements owned by lanes 0-15; pass 2 those of lanes 16-31.
  v8f acc1 = {};
  v8f acc2 = {};
#pragma unroll
  for (int kb = 0; kb < 8; ++kb) {
    const int k0 = kb * 4;
    v2f a, b1, b2;
    a[0] = lo16 ? (mrid == mrow(k0 + 0) ? 1.0f : 0.0f)
                : (mrid == mrow(k0 + 2) ? 1.0f : 0.0f);
    a[1] = lo16 ? (mrid == mrow(k0 + 1) ? 1.0f : 0.0f)
                : (mrid == mrow(k0 + 3) ? 1.0f : 0.0f);
    const float r0 = rot16f(lane, val[k0 + 0]);   // partner lane's values
    const float r1 = rot16f(lane, val[k0 + 1]);
    const float r2 = rot16f(lane, val[k0 + 2]);
    const float r3 = rot16f(lane, val[k0 + 3]);
    b1[0] = lo16 ? val[k0 + 0] : r2;              // columns = lanes 0-15's elements
    b1[1] = lo16 ? val[k0 + 1] : r3;
    b2[0] = lo16 ? r0 : val[k0 + 2];              // columns = lanes 16-31's elements
    b2[1] = lo16 ? r1 : val[k0 + 3];
    acc1 = __builtin_amdgcn_wmma_f32_16x16x4_f32(false, a, false, b1, (short)0, acc1, false, false);
    acc2 = __builtin_amdgcn_wmma_f32_16x16x4_f32(false, a, false, b2, (short)0, acc2, false, false);
  }

  // D rows 0..3 live in D-VGPRs 0..3 for lanes 0-15 (column = lane = element).
  const float resA = acc1[0] * acc1[2] + acc1[1] * acc1[3];  // valid in lanes 0-15
  const float resB = acc2[0] * acc2[2] + acc2[1] * acc2[3];  // lanes 0-15 hold lane+16's result
  const float resBs = rot16f(lane, resB);                    // move to lanes 16-31
  out[b] = lo16 ? resA : resBs;
}

extern "C" void kernel_launch(void* const* d_in, const int* in_sizes, int n_in,
                              void* d_out, int out_size, void* d_ws, size_t ws_size,
                              hipStream_t stream) {
  (void)n_in; (void)d_ws; (void)ws_size; (void)out_size;
  const float* inputs = (const float*)d_in[0];   // (131072, 8) f32
  const float* wre    = (const float*)d_in[1];   // (3,) f32
  const float* wim    = (const float*)d_in[2];   // (3,) f32
  float* out          = (float*)d_out;           // (131072,) f32

  const int batch   = in_sizes[0] / 8;
  const int threads = 256;                       // 8 wave32s per block
  const int blocks  = (batch + threads - 1) / threads;
  qring_kernel<<<blocks, threads, 0, stream>>>(inputs, wre, wim, out, batch);
}